// GCN_21543555956998
// MI455X (gfx1250) — compile-verified
//
#include <hip/hip_runtime.h>
#include <math.h>

typedef __attribute__((ext_vector_type(2))) float v2f;
typedef __attribute__((ext_vector_type(8))) float v8f;

// ---------------------------------------------------------------------------
// degree / normalization
// ---------------------------------------------------------------------------
__global__ void k_set1(float* __restrict__ p, int n) {
    int i = blockIdx.x * blockDim.x + threadIdx.x;
    if (i < n) p[i] = 1.0f;   // self-loop contributes 1 to degree
}

__global__ void k_deg(const long long* __restrict__ ei, int E, float* __restrict__ deg) {
    int e = blockIdx.x * blockDim.x + threadIdx.x;
    if (e < E) {
        int d = (int)ei[(size_t)E + e];   // dst row of edge_index
        atomicAdd(&deg[d], 1.0f);
    }
}

__global__ void k_rsqrt(float* __restrict__ p, int n) {
    int i = blockIdx.x * blockDim.x + threadIdx.x;
    if (i < n) p[i] = rsqrtf(p[i]);
}

// ---------------------------------------------------------------------------
// Layer 1 GEMM: xw1[N,64] = x[N,8] @ W1[8,64]   (V_WMMA_F32_16X16X4_F32)
// One wave per full 16-row tile (NO bounds guards -> no exec-mask ladders).
// Fused epilogue: agg1[i,f] = b1[f] + xw1[i,f]*dis[i]^2  (bias + self-loop msg)
// ---------------------------------------------------------------------------
__global__ __launch_bounds__(32) void k_gemm1_wmma(const float* __restrict__ x,
                                                   const float* __restrict__ W1,
                                                   const float* __restrict__ b1,
                                                   const float* __restrict__ dis,
                                                   float* __restrict__ xw1,
                                                   float* __restrict__ agg1) {
    const int lane  = threadIdx.x;          // 0..31
    const int row0  = blockIdx.x * 16;
    const int mrow  = row0 + (lane & 15);
    const int khalf = (lane >> 4) * 2;      // 0 for lanes 0-15, 2 for lanes 16-31
    const int ncol0 = lane & 15;

    // A tiles for k0=0 and k0=4  (lane holds K = k0+khalf, k0+khalf+1 of its row)
    v2f a0, a1;
    a0.x = x[(size_t)mrow * 8 + 0 + khalf];
    a0.y = x[(size_t)mrow * 8 + 1 + khalf];
    a1.x = x[(size_t)mrow * 8 + 4 + khalf];
    a1.y = x[(size_t)mrow * 8 + 5 + khalf];

    v8f acc[4];
    float bias[4];
#pragma unroll
    for (int nt = 0; nt < 4; ++nt) {
        const int ncol = nt * 16 + ncol0;
        v2f b0, b1v;
        b0.x  = W1[(0 + khalf) * 64 + ncol];
        b0.y  = W1[(1 + khalf) * 64 + ncol];
        b1v.x = W1[(4 + khalf) * 64 + ncol];
        b1v.y = W1[(5 + khalf) * 64 + ncol];
        v8f c = {};
        c = __builtin_amdgcn_wmma_f32_16x16x4_f32(false, a0, false, b0,  (short)0, c, false, false);
        c = __builtin_amdgcn_wmma_f32_16x16x4_f32(false, a1, false, b1v, (short)0, c, false, false);
        acc[nt]  = c;
        bias[nt] = b1[ncol];
    }

    // C/D layout: VGPR j, lanes 0-15 -> M=j, lanes 16-31 -> M=8+j; N = lane&15
    const int rbase = row0 + (lane >> 4) * 8;
#pragma unroll
    for (int j = 0; j < 8; ++j) {
        const int r  = rbase + j;
        const float s  = dis[r];
        const float s2 = s * s;
#pragma unroll
        for (int nt = 0; nt < 4; ++nt) {
            const float v = acc[nt][j];
            xw1 [(size_t)r * 64 + nt * 16 + ncol0] = v;
            agg1[(size_t)r * 64 + nt * 16 + ncol0] = bias[nt] + v * s2;
        }
    }
}

// scalar tail for rows not covered by full 16-row tiles (none when N%16==0)
__global__ void k_tail1(const float* __restrict__ x, const float* __restrict__ W1,
                        const float* __restrict__ b1, const float* __restrict__ dis,
                        float* __restrict__ xw1, float* __restrict__ agg1,
                        int rowStart, int N) {
    int t = blockIdx.x * blockDim.x + threadIdx.x;
    int total = (N - rowStart) * 64;
    if (t >= total) return;
    int r = rowStart + t / 64, f = t % 64;
    float acc = 0.0f;
    for (int k = 0; k < 8; ++k) acc += x[(size_t)r * 8 + k] * W1[k * 64 + f];
    xw1[(size_t)r * 64 + f] = acc;
    float s = dis[r];
    agg1[(size_t)r * 64 + f] = b1[f] + acc * s * s;
}

// ---------------------------------------------------------------------------
// Layer 1 edge scatter: 16 threads per edge, float4 gather + 4 f32 atomics
// ---------------------------------------------------------------------------
__global__ void k_scatter1(const long long* __restrict__ ei, int E,
                           const float* __restrict__ dis,
                           const float* __restrict__ xw1, float* __restrict__ agg) {
    int t = blockIdx.x * blockDim.x + threadIdx.x;
    if (t >= E * 16) return;
    int e = t >> 4, g = t & 15;
    int s = (int)ei[e];
    int d = (int)ei[(size_t)E + e];
    float nrm = dis[s] * dis[d];
    float4 v = ((const float4*)xw1)[(size_t)s * 16 + g];
    float* out = agg + (size_t)d * 64 + g * 4;
    atomicAdd(out + 0, v.x * nrm);
    atomicAdd(out + 1, v.y * nrm);
    atomicAdd(out + 2, v.z * nrm);
    atomicAdd(out + 3, v.w * nrm);
}

// ---------------------------------------------------------------------------
// Layer 2 GEMM: xw2[N,16] = relu(agg1)[N,64] @ W2pad[64,16]  (pad cols 12..15=0)
// ReLU fused into the A load; fused epilogue writes xw2 and
// agg2[i,f] = b2pad[f] + xw2[i,f]*dis[i]^2.  Full tiles only, no guards.
// ---------------------------------------------------------------------------
__global__ __launch_bounds__(32) void k_gemm2_wmma(const float* __restrict__ agg1,
                                                   const float* __restrict__ W2,
                                                   const float* __restrict__ b2,
                                                   const float* __restrict__ dis,
                                                   float* __restrict__ xw2,
                                                   float* __restrict__ agg2) {
    const int lane  = threadIdx.x;
    const int row0  = blockIdx.x * 16;
    const int mrow  = row0 + (lane & 15);
    const int khalf = (lane >> 4) * 2;
    const int ncol  = lane & 15;
    const bool nok  = (ncol < 12);
    const float bias = nok ? b2[ncol] : 0.0f;

    v8f c = {};
#pragma unroll
    for (int k0 = 0; k0 < 64; k0 += 4) {
        v2f a, b;
        a.x = fmaxf(agg1[(size_t)mrow * 64 + k0 + khalf + 0], 0.0f);  // relu(h1)
        a.y = fmaxf(agg1[(size_t)mrow * 64 + k0 + khalf + 1], 0.0f);
        b.x = nok ? W2[(k0 + khalf + 0) * 12 + ncol] : 0.0f;
        b.y = nok ? W2[(k0 + khalf + 1) * 12 + ncol] : 0.0f;
        c = __builtin_amdgcn_wmma_f32_16x16x4_f32(false, a, false, b, (short)0, c, false, false);
    }
    const int rbase = row0 + (lane >> 4) * 8;
#pragma unroll
    for (int j = 0; j < 8; ++j) {
        const int r  = rbase + j;
        const float s  = dis[r];
        const float v  = c[j];
        xw2 [(size_t)r * 16 + ncol] = v;
        agg2[(size_t)r * 16 + ncol] = bias + v * s * s;
    }
}

__global__ void k_tail2(const float* __restrict__ agg1, const float* __restrict__ W2,
                        const float* __restrict__ b2, const float* __restrict__ dis,
                        float* __restrict__ xw2, float* __restrict__ agg2,
                        int rowStart, int N) {
    int t = blockIdx.x * blockDim.x + threadIdx.x;
    int total = (N - rowStart) * 16;
    if (t >= total) return;
    int r = rowStart + t / 16, f = t % 16;
    float acc = 0.0f;
    if (f < 12)
        for (int k = 0; k < 64; ++k)
            acc += fmaxf(agg1[(size_t)r * 64 + k], 0.0f) * W2[k * 12 + f];
    xw2[(size_t)r * 16 + f] = acc;
    float s = dis[r];
    agg2[(size_t)r * 16 + f] = ((f < 12) ? b2[f] : 0.0f) + acc * s * s;
}

// ---------------------------------------------------------------------------
// Layer 2 edge scatter: 4 threads per edge (16 padded features / float4)
// ---------------------------------------------------------------------------
__global__ void k_scatter2(const long long* __restrict__ ei, int E,
                           const float* __restrict__ dis,
                           const float* __restrict__ xw2, float* __restrict__ agg) {
    int t = blockIdx.x * blockDim.x + threadIdx.x;
    if (t >= E * 4) return;
    int e = t >> 2, g = t & 3;
    int s = (int)ei[e];
    int d = (int)ei[(size_t)E + e];
    float nrm = dis[s] * dis[d];
    float4 v = ((const float4*)xw2)[(size_t)s * 4 + g];
    float* out = agg + (size_t)d * 16 + g * 4;
    atomicAdd(out + 0, v.x * nrm);
    atomicAdd(out + 1, v.y * nrm);
    atomicAdd(out + 2, v.z * nrm);
    atomicAdd(out + 3, v.w * nrm);
}

// ---------------------------------------------------------------------------
// global max pool (ReLU fused). Values >= 0, so uint atomicMax == IEEE max.
// ---------------------------------------------------------------------------
__global__ void k_zero(float* __restrict__ p, int n) {
    int i = blockIdx.x * blockDim.x + threadIdx.x;
    if (i < n) p[i] = 0.0f;
}

__global__ void k_pool(const long long* __restrict__ batch, const float* __restrict__ agg2,
                       unsigned* __restrict__ pooled, int N) {
    int t = blockIdx.x * blockDim.x + threadIdx.x;
    if (t >= N * 12) return;
    int i = t / 12, f = t % 12;
    float v = fmaxf(agg2[(size_t)i * 16 + f], 0.0f);   // relu(h2)
    int g = (int)batch[i];
    atomicMax(&pooled[(size_t)g * 16 + f], __float_as_uint(v));
}

// ---------------------------------------------------------------------------
// head: logits = pooled @ Wl + bl ; 2-class log_softmax
// ---------------------------------------------------------------------------
__global__ void k_head(const float* __restrict__ pooled, const float* __restrict__ Wl,
                       const float* __restrict__ bl, float* __restrict__ out, int G) {
    int g = blockIdx.x * blockDim.x + threadIdx.x;
    if (g >= G) return;
    float l0 = bl[0], l1 = bl[1];
#pragma unroll
    for (int f = 0; f < 12; ++f) {
        float p = pooled[(size_t)g * 16 + f];
        l0 += p * Wl[f * 2 + 0];
        l1 += p * Wl[f * 2 + 1];
    }
    float m   = fmaxf(l0, l1);
    float lse = m + logf(expf(l0 - m) + expf(l1 - m));
    out[g * 2 + 0] = l0 - lse;
    out[g * 2 + 1] = l1 - lse;
}

// ---------------------------------------------------------------------------
extern "C" void kernel_launch(void* const* d_in, const int* in_sizes, int n_in,
                              void* d_out, int out_size, void* d_ws, size_t ws_size,
                              hipStream_t stream) {
    const float*     x   = (const float*)d_in[0];
    const long long* ei  = (const long long*)d_in[1];
    const long long* bat = (const long long*)d_in[2];
    const float*     W1  = (const float*)d_in[3];
    const float*     b1  = (const float*)d_in[4];
    const float*     W2  = (const float*)d_in[5];
    const float*     b2  = (const float*)d_in[6];
    const float*     Wl  = (const float*)d_in[7];
    const float*     bl  = (const float*)d_in[8];

    const int N = in_sizes[0] / 8;    // 200000
    const int E = in_sizes[1] / 2;    // 1600000
    const int G = out_size / 2;       // 2000

    float* ws     = (float*)d_ws;
    float* dis    = ws;                                  // N
    float* bufA   = dis + N;                             // N*64 (xw1, later xw2+agg2)
    float* bufB   = bufA + (size_t)N * 64;               // N*64 (agg1 / h1)
    float* pooled = bufB + (size_t)N * 64;               // G*16
    float* xw2    = bufA;                                // N*16
    float* agg2   = bufA + (size_t)N * 16;               // N*16

    const int B = 256;
    float* out = (float*)d_out;

    const int fullTiles = N / 16;          // exact-tile WMMA dispatch
    const int tailStart = fullTiles * 16;  // == N when N%16==0
    const int tailRows  = N - tailStart;

    // degree + rsqrt
    k_set1 <<<(N + B - 1) / B, B, 0, stream>>>(dis, N);
    k_deg  <<<(E + B - 1) / B, B, 0, stream>>>(ei, E, dis);
    k_rsqrt<<<(N + B - 1) / B, B, 0, stream>>>(dis, N);

    // layer 1: WMMA GEMM with fused bias+self-loop epilogue, then edge scatter
    k_gemm1_wmma<<<fullTiles, 32, 0, stream>>>(x, W1, b1, dis, bufA, bufB);
    if (tailRows > 0)
        k_tail1<<<(tailRows * 64 + B - 1) / B, B, 0, stream>>>(x, W1, b1, dis, bufA, bufB,
                                                               tailStart, N);
    k_scatter1<<<((size_t)E * 16 + B - 1) / B, B, 0, stream>>>(ei, E, dis, bufA, bufB);

    // layer 2: relu fused into A-load; fused bias+self-loop epilogue; scatter
    k_gemm2_wmma<<<fullTiles, 32, 0, stream>>>(bufB, W2, b2, dis, xw2, agg2);
    if (tailRows > 0)
        k_tail2<<<(tailRows * 16 + B - 1) / B, B, 0, stream>>>(bufB, W2, b2, dis, xw2, agg2,
                                                               tailStart, N);
    k_scatter2<<<((size_t)E * 4 + B - 1) / B, B, 0, stream>>>(ei, E, dis, xw2, agg2);

    // pool (relu fused) + head
    k_zero<<<(G * 16 + B - 1) / B, B, 0, stream>>>(pooled, G * 16);
    k_pool<<<((size_t)N * 12 + B - 1) / B, B, 0, stream>>>(bat, agg2, (unsigned*)pooled, N);
    k_head<<<(G + B - 1) / B, B, 0, stream>>>(pooled, Wl, bl, out, G);
}